// M2MRF_21139829031565
// MI455X (gfx1250) — compile-verified
//
#include <hip/hip_runtime.h>

typedef float v2f __attribute__((ext_vector_type(2)));
typedef float v8f __attribute__((ext_vector_type(8)));

#define WMMA_F32(a_, b_, c_) \
  __builtin_amdgcn_wmma_f32_16x16x4_f32(false, (a_), false, (b_), (short)0, (c_), false, false)

// ---------------------------------------------------------------------------
// Problem constants
//   x:[4,256,256,256]  enc 64ch  patches 8x8 (32x32 grid, 1024/img, 4096 total)
//   out:[4,256,128,128]
// ---------------------------------------------------------------------------
#define NB      4
#define CIN     256
#define HH      256
#define WW      256
#define EC      64
#define GW      32
#define LPERN   1024
#define LTOT    4096
#define K1      4096      // EC*64 unfold vector
#define COUT    256

// workspace layout (floats)
#define ENC_FLOATS  (16777216)   // [4096 patches][4096]  64 MiB
#define T_FLOATS    (262144)     // [64 f][4096 l]         1 MiB
#define M_FLOATS    (262144)     // [16 s][256 o][64 f]    1 MiB

// LDS pad strides (floats)
//  A buffers: lane stride % 64 == 4 -> b64 A-loads hit all 64 banks once.
//  B buffers: K-pair interleaved [kpair][col][2]; pair-row stride % 64 == 32
//  so half-wave 1 (K+2) lands on banks 32..63 while half-wave 0 covers 0..31.
#define WS_STRIDE   260   // A: 256-col weight chunks
#define MS_STRIDE   68    // A: 64-col M_s
#define XS2_STRIDE  288   // B: encode, 128 cols * 2 (288 % 64 == 32)
#define ES2_STRIDE  96    // B: fc1/decode, 32 cols * 2 (96 % 64 == 32)

static __device__ __forceinline__ v8f v8f_zero() {
  v8f z = {0.f, 0.f, 0.f, 0.f, 0.f, 0.f, 0.f, 0.f};
  return z;
}

// ---------------------------------------------------------------------------
// Kernel A: fold FC2 + decode weights:  M_s[o][f] = sum_e Wdec[o,e]*W2[e*16+s,f]
//           cb[s][o] = sum_e Wdec[o,e]*b2[e*16+s] + bdec[o]
// ---------------------------------------------------------------------------
__global__ __launch_bounds__(256) void k_fuse(
    const float* __restrict__ W2, const float* __restrict__ b2,
    const float* __restrict__ Wdec, const float* __restrict__ bdec,
    float* __restrict__ Mf, float* __restrict__ cb) {
  int idx = blockIdx.x * 256 + threadIdx.x;   // 0 .. 262143
  int s = idx >> 14;
  int o = (idx >> 6) & 255;
  int f = idx & 63;
  float sum = 0.f, sb = 0.f;
  #pragma unroll 8
  for (int e = 0; e < 64; ++e) {
    float wd = Wdec[o * 64 + e];
    sum += wd * W2[(e * 16 + s) * 64 + f];
    sb  += wd * b2[e * 16 + s];
  }
  Mf[idx] = sum;
  if (f == 0) cb[s * 256 + o] = sb + bdec[o];
}

// ---------------------------------------------------------------------------
// Kernel B: encode GEMM  enc[l][k] = (Wenc(64x256) * x) written patch-major.
// Block = (n, gh, gw-pair): 128 pixels = 2 patches; 8 waves = 4(M) x 2(N) tiles.
// ---------------------------------------------------------------------------
__global__ __launch_bounds__(256) void k_encode(
    const float* __restrict__ x, const float* __restrict__ Wenc,
    const float* __restrict__ benc, float* __restrict__ enc) {
  extern __shared__ float lds[];
  float* ws = lds;                       // [64][WS_STRIDE] full Wenc (A)
  float* xs = lds + 64 * WS_STRIDE;      // [32 kpair][XS2_STRIDE] x chunk (B)

  const int tid = threadIdx.x;
  const int b   = blockIdx.x;
  const int n   = b >> 9;
  const int rem = b & 511;
  const int gh  = rem >> 4;
  const int gw0 = (rem & 15) * 2;

  // stage all of Wenc (64x256) into LDS
  {
    const float4* src = (const float4*)Wenc;
    #pragma unroll
    for (int i = 0; i < 16; ++i) {
      int v = i * 256 + tid;             // vec4 index, 4096 total
      float4 d = src[v];
      int e = v >> 6;
      int k = (v & 63) * 4;
      *(float4*)&ws[e * WS_STRIDE + k] = d;
    }
  }

  const int lane = tid & 31;
  const int wv   = tid >> 5;
  const int lm   = lane & 15;
  const int lh   = lane >> 4;
  const int e0   = (wv >> 1) * 16;       // M band
  const int c0   = (wv & 1) * 64;        // N band (4 x 16 sub-tiles)

  v8f acc[4];
  #pragma unroll
  for (int t = 0; t < 4; ++t) acc[t] = v8f_zero();

  // staging assignment: thread -> (k-pair, kh); two adjacent channels per thread
  const int kk2 = tid >> 3;              // 0..31
  const int kh  = tid & 7;

  for (int kc = 0; kc < 256; kc += 64) {
    __syncthreads();
    {
      const float* rowA = x +
          ((((size_t)n * CIN + (kc + 2 * kk2)) * HH + (gh * 8 + kh)) * WW + gw0 * 8);
      const float* rowB = rowA + (size_t)HH * WW;   // next channel
      float4 A4[4], B4[4];
      #pragma unroll
      for (int q = 0; q < 4; ++q) {
        A4[q] = ((const float4*)rowA)[q];
        B4[q] = ((const float4*)rowB)[q];
      }
      const float* fa = (const float*)A4;
      const float* fb = (const float*)B4;
      #pragma unroll
      for (int i = 0; i < 16; ++i) {
        int cc = (i >> 3) * 64 + kh * 8 + (i & 7);  // patch-major column
        v2f t; t.x = fa[i]; t.y = fb[i];            // (K even, K odd) pair
        *(v2f*)&xs[kk2 * XS2_STRIDE + cc * 2] = t;
      }
    }
    __syncthreads();

    #pragma unroll 4
    for (int kk = 0; kk < 64; kk += 4) {
      int ka = kk + 2 * lh;                          // K = 2*half + vgpr
      v2f a = *(const v2f*)&ws[(e0 + lm) * WS_STRIDE + kc + ka];
      #pragma unroll
      for (int t = 0; t < 4; ++t) {
        v2f bb = *(const v2f*)&xs[(ka >> 1) * XS2_STRIDE + (c0 + t * 16 + lm) * 2];
        acc[t] = WMMA_F32(a, bb, acc[t]);
      }
    }
  }

  // write patch-major: enc[(n*1024+l)*4096 + e*64 + ph]; 64B-contiguous per half-row
  size_t base_l = (size_t)n * LPERN + gh * GW + gw0;
  #pragma unroll
  for (int t = 0; t < 4; ++t) {
    int c  = c0 + t * 16 + lm;
    int pp = c >> 6;
    int ph = c & 63;
    #pragma unroll
    for (int r = 0; r < 8; ++r) {
      int e = e0 + r + 8 * lh;
      enc[(base_l + pp) * (size_t)K1 + e * 64 + ph] = acc[t][r] + benc[e];
    }
  }
}

// ---------------------------------------------------------------------------
// Kernel C1: FC1  T[f][l] = W1(64x4096) * enc + b1.  Block = 32 patches.
// ---------------------------------------------------------------------------
__global__ __launch_bounds__(256) void k_fc1(
    const float* __restrict__ enc, const float* __restrict__ W1,
    const float* __restrict__ b1, float* __restrict__ Tbuf) {
  extern __shared__ float lds[];
  float* w1s = lds;                      // [64][WS_STRIDE] K-chunk of W1 (A)
  float* es  = lds + 64 * WS_STRIDE;     // [128 kpair][ES2_STRIDE] E chunk (B)

  const int tid  = threadIdx.x;
  const int l0   = blockIdx.x * 32;
  const int lane = tid & 31, wv = tid >> 5;
  const int lm   = lane & 15, lh = lane >> 4;
  const int f0   = (wv >> 1) * 16;
  const int c0   = (wv & 1) * 16;

  v8f acc = v8f_zero();

  for (int kc = 0; kc < K1; kc += 256) {
    __syncthreads();
    // stage W1 chunk (coalesced float4)
    #pragma unroll
    for (int i = 0; i < 16; ++i) {
      int v  = i * 256 + tid;            // 0..4095 vec4 units
      int f  = v >> 6;
      int kq = (v & 63) * 4;
      float4 d = *(const float4*)(W1 + (size_t)f * K1 + kc + kq);
      *(float4*)&w1s[f * WS_STRIDE + kq] = d;
    }
    // stage E chunk, transposing to K-pair-interleaved [kpair][patch][2]
    #pragma unroll
    for (int i = 0; i < 8; ++i) {
      int v  = i * 256 + tid;            // 0..2047
      int c  = v >> 6;
      int kq = (v & 63) * 4;
      float4 d = *(const float4*)(enc + (size_t)(l0 + c) * K1 + kc + kq);
      v2f t0; t0.x = d.x; t0.y = d.y;
      v2f t1; t1.x = d.z; t1.y = d.w;
      *(v2f*)&es[((kq >> 1) + 0) * ES2_STRIDE + c * 2] = t0;
      *(v2f*)&es[((kq >> 1) + 1) * ES2_STRIDE + c * 2] = t1;
    }
    __syncthreads();

    #pragma unroll 4
    for (int kk = 0; kk < 256; kk += 4) {
      int ka = kk + 2 * lh;
      v2f a  = *(const v2f*)&w1s[(f0 + lm) * WS_STRIDE + ka];
      v2f bb = *(const v2f*)&es[(ka >> 1) * ES2_STRIDE + (c0 + lm) * 2];
      acc = WMMA_F32(a, bb, acc);
    }
  }

  #pragma unroll
  for (int r = 0; r < 8; ++r) {
    int f = f0 + r + 8 * lh;
    Tbuf[(size_t)f * LTOT + l0 + c0 + lm] = acc[r] + b1[f];
  }
}

// ---------------------------------------------------------------------------
// Kernel C2: fused FC2+fold+decode:  out = M_s * T + cb.
// Block = 32 patches x 4 s-values; 512 blocks.
// ---------------------------------------------------------------------------
__global__ __launch_bounds__(256) void k_decode(
    const float* __restrict__ Tbuf, const float* __restrict__ Mf,
    const float* __restrict__ cb, float* __restrict__ out) {
  extern __shared__ float lds[];
  float* ms = lds;                       // [256][MS_STRIDE] one M_s (A)
  float* ts = lds + 256 * MS_STRIDE;     // [32 kpair][ES2_STRIDE] T tile (B)

  const int tid = threadIdx.x;
  const int pb  = blockIdx.x >> 2;       // patch-tile 0..127
  const int sg  = blockIdx.x & 3;        // s group
  const int l0  = pb * 32;
  const int n   = l0 >> 10;
  const int gh  = (l0 & 1023) >> 5;      // block = one full patch row (gw 0..31)

  // stage T tile [64][32] as K-pair interleaved
  #pragma unroll
  for (int i = 0; i < 4; ++i) {
    int v  = i * 256 + tid;              // 0..1023 pairs
    int f2 = v >> 5;                     // 0..31
    int c  = v & 31;
    v2f t;
    t.x = Tbuf[(size_t)(2 * f2 + 0) * LTOT + l0 + c];
    t.y = Tbuf[(size_t)(2 * f2 + 1) * LTOT + l0 + c];
    *(v2f*)&ts[f2 * ES2_STRIDE + c * 2] = t;
  }

  const int lane = tid & 31, wv = tid >> 5;
  const int lm   = lane & 15, lh = lane >> 4;

  for (int ss = 0; ss < 4; ++ss) {
    int s = sg * 4 + ss;
    __syncthreads();                     // ts ready / previous ms consumers done
    #pragma unroll
    for (int i = 0; i < 16; ++i) {
      int v  = i * 256 + tid;            // vec4 idx 0..4095
      int o  = v >> 4;
      int fq = (v & 15) * 4;
      float4 d = *(const float4*)(Mf + (size_t)s * 16384 + o * 64 + fq);
      *(float4*)&ms[o * MS_STRIDE + fq] = d;
    }
    __syncthreads();

    int sy = s >> 2, sx = s & 3;
    #pragma unroll
    for (int j = 0; j < 4; ++j) {
      int tt = wv * 4 + j;               // 32 tiles / 8 waves
      int o0 = (tt >> 1) * 16;
      int ct = (tt & 1) * 16;
      v8f acc = v8f_zero();
      #pragma unroll
      for (int kk = 0; kk < 64; kk += 4) {
        int ka = kk + 2 * lh;
        v2f a  = *(const v2f*)&ms[(o0 + lm) * MS_STRIDE + ka];
        v2f bb = *(const v2f*)&ts[(ka >> 1) * ES2_STRIDE + (ct + lm) * 2];
        acc = WMMA_F32(a, bb, acc);
      }
      int gw = ct + lm;
      int h2 = gh * 4 + sy;
      int w2 = gw * 4 + sx;
      #pragma unroll
      for (int r = 0; r < 8; ++r) {
        int o = o0 + r + 8 * lh;
        out[(((size_t)n * COUT + o) * 128 + h2) * 128 + w2] = acc[r] + cb[s * 256 + o];
      }
    }
  }
}

// ---------------------------------------------------------------------------
extern "C" void kernel_launch(void* const* d_in, const int* in_sizes, int n_in,
                              void* d_out, int out_size, void* d_ws, size_t ws_size,
                              hipStream_t stream) {
  const float* x    = (const float*)d_in[0];
  const float* Wenc = (const float*)d_in[1];
  const float* benc = (const float*)d_in[2];
  const float* W1   = (const float*)d_in[3];
  const float* b1   = (const float*)d_in[4];
  const float* W2   = (const float*)d_in[5];
  const float* b2   = (const float*)d_in[6];
  const float* Wdec = (const float*)d_in[7];
  const float* bdec = (const float*)d_in[8];
  float* out = (float*)d_out;

  float* ws   = (float*)d_ws;
  float* enc  = ws;                                  // 64 MiB
  float* Tbuf = ws + ENC_FLOATS;                     // 1 MiB
  float* Mf   = Tbuf + T_FLOATS;                     // 1 MiB
  float* cbuf = Mf + M_FLOATS;                       // 16 KiB

  k_fuse<<<1024, 256, 0, stream>>>(W2, b2, Wdec, bdec, Mf, cbuf);

  size_t ldsB = (size_t)(64 * WS_STRIDE + 32 * XS2_STRIDE) * sizeof(float);
  k_encode<<<2048, 256, ldsB, stream>>>(x, Wenc, benc, enc);

  size_t ldsC1 = (size_t)(64 * WS_STRIDE + 128 * ES2_STRIDE) * sizeof(float);
  k_fc1<<<128, 256, ldsC1, stream>>>(enc, W1, b1, Tbuf);

  size_t ldsC2 = (size_t)(256 * MS_STRIDE + 32 * ES2_STRIDE) * sizeof(float);
  k_decode<<<512, 256, ldsC2, stream>>>(Tbuf, Mf, cbuf, out);
}